// NURBSSurface_19722489823360
// MI455X (gfx1250) — compile-verified
//
#include <hip/hip_runtime.h>
#include <stdint.h>

// NURBS surface eval, degree 3x3, 256x256 control net, NDER=1.
// CDNA5 (gfx1250) specifics used:
//  - wave32 / 8-wave blocks of 256
//  - global_load_async_to_lds_b32 double-buffering of eval points (ASYNCcnt)
//  - s_wait_asynccnt split-counter waits
//  - v_rcp_f32 via __builtin_amdgcn_rcpf (avoids ~24 full f32 div expansions/point)
//  - non-temporal b128 stores for the streamed 32MB output
// WMMA is intentionally NOT used: both operands of the 16x4 contraction are
// per-lane data (per-point basis weights AND per-point gathered patch), which
// cannot map onto a wave-shared WMMA matrix without >8x wasted MACs plus
// cross-lane marshaling; plain FMAs are strictly faster here.

#define TILE 256
#define NCPT 256

typedef float v4f __attribute__((ext_vector_type(4)));

__device__ __forceinline__ float knotv(int i) {
  // clamped uniform knot vector: zeros(3), linspace(0,1,254), ones(3)
  float t = (float)(i - 3) * (1.0f / 253.0f);
  t = t < 0.0f ? 0.0f : t;
  t = t > 1.0f ? 1.0f : t;
  return t;
}

__device__ __forceinline__ int find_span(float u) {
  int j = (int)(u * 253.0f);          // u in [0,1)
  j = j < 0 ? 0 : (j > 252 ? 252 : j);
  int span = j + 3;
  if (u < knotv(span))            span -= 1;
  else if (u >= knotv(span + 1))  span += 1;
  span = span < 3 ? 3 : (span > 255 ? 255 : span);
  const float kend = 252.0f / 253.0f; // knots[n_cols-1]
  if (fabsf(u - kend) <= 1e-5f + 1e-5f * kend) span = 255;
  return span;
}

// Cox-de Boor basis + first derivative, degree 3, fully unrolled in registers.
__device__ __forceinline__ void basis_d1(float u, int span, float* N0, float* N1) {
  float left[4], right[4];
  float ndu[4][4];
  float inv3[3];
  ndu[0][0] = 1.0f;
#pragma unroll
  for (int j = 1; j <= 3; ++j) {
    left[j]  = u - knotv(span + 1 - j);
    right[j] = knotv(span + j) - u;
    float saved = 0.0f;
#pragma unroll
    for (int r = 0; r < j; ++r) {
      ndu[j][r] = right[r + 1] + left[j - r];
      float inv = __builtin_amdgcn_rcpf(ndu[j][r]);
      if (j == 3) inv3[r] = inv;            // reused by derivative below
      float temp = ndu[r][j - 1] * inv;
      ndu[r][j] = fmaf(right[r + 1], temp, saved);
      saved = left[j - r] * temp;
    }
    ndu[j][j] = saved;
  }
#pragma unroll
  for (int r = 0; r < 4; ++r) N0[r] = ndu[r][3];
#pragma unroll
  for (int r = 0; r < 4; ++r) {
    float d = 0.0f;
    if (r >= 1) d += ndu[r - 1][2] * inv3[r - 1];
    if (r <= 2) d -= ndu[r][2] * inv3[r];
    N1[r] = 3.0f * d;
  }
}

__global__ __launch_bounds__(TILE) void nurbs_eval_kernel(
    const float* __restrict__ px, const float* __restrict__ py,
    const float* __restrict__ cp, float* __restrict__ out,
    int n, int tiles) {
  __shared__ float sx[2][TILE];
  __shared__ float sy[2][TILE];
  const int tid = threadIdx.x;
  float* __restrict__ out00 = out;                      // [N,4] d00
  float* __restrict__ outnr = out + (size_t)n * 4;      // [N,4] normals

  // Async-copy one tile of eval points into an LDS buffer (per-lane b32).
  auto issue_tile = [&](int buf, long t) {
    long i = t * TILE + tid;
    if (i >= n) i = n - 1;                              // safe clamp for tail
    unsigned ldx = (unsigned)(uintptr_t)&sx[buf][tid];  // low32 == LDS offset
    unsigned ldy = (unsigned)(uintptr_t)&sy[buf][tid];
    unsigned long long gx = (unsigned long long)(uintptr_t)(px + i);
    unsigned long long gy = (unsigned long long)(uintptr_t)(py + i);
    asm volatile("global_load_async_to_lds_b32 %0, %1, off"
                 :: "v"(ldx), "v"(gx) : "memory");
    asm volatile("global_load_async_to_lds_b32 %0, %1, off"
                 :: "v"(ldy), "v"(gy) : "memory");
  };

  long t0 = blockIdx.x;
  if (t0 < tiles) issue_tile(0, t0);

  int buf = 0;
  for (long t = t0; t < tiles; t += gridDim.x, buf ^= 1) {
    long tn = t + (long)gridDim.x;
    if (tn < tiles) {
      issue_tile(buf ^ 1, tn);                          // prefetch next tile
      asm volatile("s_wait_asynccnt 2" ::: "memory");   // current tile landed
    } else {
      asm volatile("s_wait_asynccnt 0" ::: "memory");
    }

    float u = sx[buf][tid];
    float v = sy[buf][tid];
    long i = t * TILE + tid;
    if (i < n) {
      int spx = find_span(u);
      int spy = find_span(v);
      float Nx0[4], Nx1[4], Ny0[4], Ny1[4];
      basis_d1(u, spx, Nx0, Nx1);
      basis_d1(v, spy, Ny0, Ny1);

      int bx = spx - 3, by = spy - 3;

      // Stage 1: contract over r (x direction) for value and x-derivative.
      float A[4][3], B[4][3];
#pragma unroll
      for (int s = 0; s < 4; ++s)
#pragma unroll
        for (int c = 0; c < 3; ++c) { A[s][c] = 0.0f; B[s][c] = 0.0f; }

#pragma unroll
      for (int r = 0; r < 4; ++r) {
        const float* row = cp + ((size_t)(bx + r) * NCPT + (size_t)by) * 3;
        float p[12];
#pragma unroll
        for (int q = 0; q < 12; ++q) p[q] = row[q];     // 48B row gather
        float w0 = Nx0[r], w1 = Nx1[r];
#pragma unroll
        for (int s = 0; s < 4; ++s)
#pragma unroll
          for (int c = 0; c < 3; ++c) {
            float pv = p[s * 3 + c];
            A[s][c] = fmaf(w0, pv, A[s][c]);
            B[s][c] = fmaf(w1, pv, B[s][c]);
          }
      }

      // Stage 2: contract over s (y direction).
      float d00[3] = {0, 0, 0}, d10[3] = {0, 0, 0}, d01[3] = {0, 0, 0};
#pragma unroll
      for (int s = 0; s < 4; ++s) {
        float y0 = Ny0[s], y1 = Ny1[s];
#pragma unroll
        for (int c = 0; c < 3; ++c) {
          d00[c] = fmaf(y0, A[s][c], d00[c]);
          d10[c] = fmaf(y0, B[s][c], d10[c]);
          d01[c] = fmaf(y1, A[s][c], d01[c]);
        }
      }
      // Homogeneous w channel: patch w==1 -> (sum Nx0)*(sum Ny0).
      float sx0 = (Nx0[0] + Nx0[1]) + (Nx0[2] + Nx0[3]);
      float sy0 = (Ny0[0] + Ny0[1]) + (Ny0[2] + Ny0[3]);
      float d00w = sx0 * sy0;

      // normal = normalize(cross(d10, d01))
      float nx = fmaf(d10[1], d01[2], -d10[2] * d01[1]);
      float ny = fmaf(d10[2], d01[0], -d10[0] * d01[2]);
      float nz = fmaf(d10[0], d01[1], -d10[1] * d01[0]);
      float len = sqrtf(fmaf(nx, nx, fmaf(ny, ny, nz * nz)));
      float invl = __builtin_amdgcn_rcpf(fmaxf(len, 1e-12f));

      v4f o0 = {d00[0], d00[1], d00[2], d00w};
      v4f o1 = {nx * invl, ny * invl, nz * invl, 0.0f};
      __builtin_nontemporal_store(o0, (v4f*)(out00 + (size_t)i * 4));
      __builtin_nontemporal_store(o1, (v4f*)(outnr + (size_t)i * 4));
    }
  }
}

extern "C" void kernel_launch(void* const* d_in, const int* in_sizes, int n_in,
                              void* d_out, int out_size, void* d_ws, size_t ws_size,
                              hipStream_t stream) {
  const float* px = (const float*)d_in[0];
  const float* py = (const float*)d_in[1];
  const float* cp = (const float*)d_in[2];
  float* out = (float*)d_out;
  int n = in_sizes[0];                       // 1,048,576
  int tiles = (n + TILE - 1) / TILE;         // 4096
  int blocks = tiles < 1024 ? tiles : 1024;  // 4 tiles/block -> real prefetch
  nurbs_eval_kernel<<<blocks, TILE, 0, stream>>>(px, py, cp, out, n, tiles);
}